// HypergraphRayleighQuotientLossGeneralized_63745904607410
// MI455X (gfx1250) — compile-verified
//
#include <hip/hip_runtime.h>
#include <hip/hip_bf16.h>

// ---------------------------------------------------------------------------
// Hypergraph Rayleigh quotient loss, gfx1250 (CDNA5, wave32).
// Column reductions use V_WMMA_F32_16X16X4_F32 with an all-ones A matrix.
// ---------------------------------------------------------------------------

typedef __attribute__((ext_vector_type(2))) float v2f;
typedef __attribute__((ext_vector_type(8))) float v8f;

#define KDIM 16
#define EPSF 1e-6f

// ---- zero workspace --------------------------------------------------------
__global__ void hg_zero_kernel(float* __restrict__ p, long long n) {
    long long i = (long long)blockIdx.x * blockDim.x + threadIdx.x;
    if (i < n) p[i] = 0.0f;
}

// ---- degree counting: Dv[node]++, De[edge]++ -------------------------------
__global__ void hg_degree_kernel(const long long* __restrict__ idx,
                                 float* __restrict__ Dv, float* __restrict__ De,
                                 long long nnz) {
    long long i = (long long)blockIdx.x * blockDim.x + threadIdx.x;
    if (i < nnz) {
        long long node = idx[i];
        long long edge = idx[nnz + i];
        atomicAdd(&Dv[node], 1.0f);
        atomicAdd(&De[edge], 1.0f);
    }
}

// ---- nZ = clip(Dv,eps)^(-1/2) * Z ------------------------------------------
__global__ void hg_norm_kernel(const float* __restrict__ Z,
                               const float* __restrict__ Dv,
                               float* __restrict__ nZ, long long total) {
    long long i = (long long)blockIdx.x * blockDim.x + threadIdx.x;
    if (i < total) {
        long long node = i >> 4;
        float d = fmaxf(Dv[node], EPSF);
        nZ[i] = Z[i] * (1.0f / sqrtf(d));
    }
}

// ---- scatter: S[edge,k] += nZ[node,k]  (16 threads per pin) ----------------
__global__ void hg_scatter_kernel(const long long* __restrict__ idx,
                                  const float* __restrict__ nZ,
                                  float* __restrict__ S, long long nnz) {
    long long i = (long long)blockIdx.x * blockDim.x + threadIdx.x;
    if (i < nnz * KDIM) {
        long long pin = i >> 4;
        int k = (int)(i & 15);
        long long node = idx[pin];
        long long edge = idx[nnz + pin];
        atomicAdd(&S[edge * KDIM + k], nZ[node * KDIM + k]);
    }
}

// ---- WMMA column-sum reduction --------------------------------------------
// MODE 0: out[k] += sum_r vals[r,k]^2 / clip(deg[r],eps)        (theta)
// MODE 1: out[k] += sum_r vals[r,k]^2 * clip(deg[r],eps)        (f^T Dv f)
// One wave reduces groups of 4 rows per V_WMMA_F32_16X16X4_F32:
//   A = ones(16x4), B = 4x16 tile of transformed values, C accumulates.
// Wave/group indices pass through readfirstlane -> scalar loop control, so
// EXEC is all-ones at every WMMA (guarded loads only predicate the loads).
template <int MODE>
__global__ void hg_colsum_wmma_kernel(const float* __restrict__ vals,
                                      const float* __restrict__ deg,
                                      float* __restrict__ out,
                                      int rows) {
    const int lane = (int)(threadIdx.x & 31);
    // uniform by construction; force into SGPRs for scalar loop control
    const int wave = __builtin_amdgcn_readfirstlane(
        ((int)blockIdx.x * (int)blockDim.x + (int)threadIdx.x) >> 5);
    const int nwaves = ((int)gridDim.x * (int)blockDim.x) >> 5;
    const int col  = lane & 15;
    const int half = lane >> 4;     // lanes 0-15 -> rows {r,r+1}; 16-31 -> {r+2,r+3}

    v2f a; a.x = 1.0f; a.y = 1.0f;  // all-ones 16x4 A matrix
    v8f c = {};

    const int ngroups = (rows + 3) >> 2;
    for (int g = wave; g < ngroups; g += nwaves) {
        const int r0 = (g << 2) + half * 2;
        const int r1 = r0 + 1;
        float v0 = 0.0f, v1 = 0.0f;
        if (r0 < rows) {
            float x = vals[(long long)r0 * KDIM + col];
            float d = fmaxf(deg[r0], EPSF);
            v0 = (MODE == 0) ? (x * x) / d : (x * x) * d;
        }
        if (r1 < rows) {
            float x = vals[(long long)r1 * KDIM + col];
            float d = fmaxf(deg[r1], EPSF);
            v1 = (MODE == 0) ? (x * x) / d : (x * x) * d;
        }
        v2f b; b.x = v0; b.y = v1;
        // D = ones(16x4) x B(4x16) + C : every row of D = column sums of B tile
        c = __builtin_amdgcn_wmma_f32_16x16x4_f32(
                /*neg_a=*/false, a, /*neg_b=*/false, b,
                /*c_mod=*/(short)0, c, /*reuse_a=*/false, /*reuse_b=*/false);
    }
    // C/D layout: VGPR0, lanes 0-15 hold D[M=0, N=lane] = column sums
    if (lane < 16) atomicAdd(&out[lane], c[0]);
}

// ---- separation stats on f = Z[:,0] ----------------------------------------
__global__ void hg_sepstats_kernel(const float* __restrict__ Z,
                                   float* __restrict__ sepbuf, int n) {
    __shared__ float red0[256];
    __shared__ float red1[256];
    __shared__ float red2[256];
    __shared__ float red3[256];
    int tid = threadIdx.x;
    int i = (int)blockIdx.x * (int)blockDim.x + tid;
    float f = (i < n) ? Z[(long long)i * KDIM] : 0.0f;
    red0[tid] = (f > 0.0f) ? f : 0.0f;
    red1[tid] = (f < 0.0f) ? f : 0.0f;
    red2[tid] = (f > 0.0f) ? 1.0f : 0.0f;
    red3[tid] = (f < 0.0f) ? 1.0f : 0.0f;
    __syncthreads();
    for (int s = 128; s > 0; s >>= 1) {
        if (tid < s) {
            red0[tid] += red0[tid + s];
            red1[tid] += red1[tid + s];
            red2[tid] += red2[tid + s];
            red3[tid] += red3[tid + s];
        }
        __syncthreads();
    }
    if (tid == 0) {
        atomicAdd(&sepbuf[0], red0[0]);   // pos sum
        atomicAdd(&sepbuf[1], red1[0]);   // neg sum
        atomicAdd(&sepbuf[2], red2[0]);   // pos count
        atomicAdd(&sepbuf[3], red3[0]);   // neg count
    }
}

// ---- finalize scalar -------------------------------------------------------
__global__ void hg_finalize_kernel(const float* __restrict__ theta,
                                   const float* __restrict__ fDvf,
                                   const float* __restrict__ sepbuf,
                                   float* __restrict__ out) {
    if (threadIdx.x == 0 && blockIdx.x == 0) {
        float acc = 0.0f;
        for (int k = 0; k < KDIM; ++k) {
            float den = fmaxf(fDvf[k], EPSF);
            float rq = 1.0f - theta[k] / den;
            if (!isfinite(rq)) rq = 0.0f;
            acc += rq;
        }
        float rayleigh = acc / (float)KDIM;
        float pc = sepbuf[2], nc = sepbuf[3];
        float pm = sepbuf[0] / fmaxf(pc, 1.0f);
        float nm = sepbuf[1] / fmaxf(nc, 1.0f);
        float sep = fabsf(pm - nm);
        float pen = ((pc == 0.0f) || (nc == 0.0f)) ? 1.0f : 1.0f / (sep + EPSF);
        out[0] = rayleigh + 0.1f * pen;
    }
}

// ---------------------------------------------------------------------------
extern "C" void kernel_launch(void* const* d_in, const int* in_sizes, int n_in,
                              void* d_out, int out_size, void* d_ws, size_t ws_size,
                              hipStream_t stream) {
    (void)n_in; (void)out_size; (void)ws_size;

    const float* Z = (const float*)d_in[0];
    const long long* idx = (const long long*)d_in[1];   // int64 (2, NNZ), row 0 = node, row 1 = edge

    const long long num_nodes = (long long)in_sizes[0] / KDIM;
    const long long nnz       = (long long)in_sizes[1] / 2;
    const long long num_edges = 100000;                 // module constant in reference

    float* ws     = (float*)d_ws;
    float* Dv     = ws;                                 // num_nodes
    float* De     = Dv + num_nodes;                     // num_edges
    float* nZ     = De + num_edges;                     // num_nodes*K
    float* S      = nZ + num_nodes * KDIM;              // num_edges*K
    float* theta  = S + num_edges * KDIM;               // K
    float* fDvf   = theta + KDIM;                       // K
    float* sepbuf = fDvf + KDIM;                        // 4
    const long long ws_elems = (sepbuf + 4) - ws;

    const int BS = 256;

    hg_zero_kernel<<<(int)((ws_elems + BS - 1) / BS), BS, 0, stream>>>(ws, ws_elems);

    hg_degree_kernel<<<(int)((nnz + BS - 1) / BS), BS, 0, stream>>>(idx, Dv, De, nnz);

    const long long nk = num_nodes * KDIM;
    hg_norm_kernel<<<(int)((nk + BS - 1) / BS), BS, 0, stream>>>(Z, Dv, nZ, nk);

    const long long scat = nnz * KDIM;
    hg_scatter_kernel<<<(int)((scat + BS - 1) / BS), BS, 0, stream>>>(idx, nZ, S, nnz);

    // theta[k] = sum_e S[e,k]^2 / clip(De[e])
    const int egroups = (int)((num_edges + 3) / 4);
    hg_colsum_wmma_kernel<0><<<(egroups * 32 + BS - 1) / BS, BS, 0, stream>>>(
        S, De, theta, (int)num_edges);

    // fDvf[k] = sum_n Z[n,k]^2 * clip(Dv[n])
    const int ngroups = (int)((num_nodes + 3) / 4);
    hg_colsum_wmma_kernel<1><<<(ngroups * 32 + BS - 1) / BS, BS, 0, stream>>>(
        Z, Dv, fDvf, (int)num_nodes);

    hg_sepstats_kernel<<<(int)((num_nodes + BS - 1) / BS), BS, 0, stream>>>(
        Z, sepbuf, (int)num_nodes);

    hg_finalize_kernel<<<1, 64, 0, stream>>>(theta, fDvf, sepbuf, (float*)d_out);
}